// DetectionLoss_32212254720316
// MI455X (gfx1250) — compile-verified
//
#include <hip/hip_runtime.h>
#include <hip/hip_bf16.h>
#include <math.h>

// ---- detection-loss constants (match reference) ----
#define POS_T  0.5f
#define NEG_T  0.4f
#define ALPHA_ 0.25f
#define BETA_  1.0f
#define EPS_   1e-6f

#define TPB   256   // 8 wave32s per block
#define MAXM  64    // gt boxes per batch (LDS staging capacity)

// Kernel 1: fused IoU match + focal + smooth-L1, per-block partial sums.
// grid = (ceil(N/TPB), B). partials[blockFlat*4 + {0..3}] = {focal, valid, sl1, pos}
__global__ __launch_bounds__(TPB) void det_loss_main(
    const float* __restrict__ cls,      // (B,N,1)
    const float* __restrict__ breg,     // (B,N,4)
    const float* __restrict__ gt,       // (B,M,4) corners x1,y1,x2,y2
    const float* __restrict__ anchors,  // (N,4)   cx,cy,w,h
    float* __restrict__ partials,
    int N, int M)
{
    __shared__ float4 s_gt[MAXM];
    __shared__ float  red[4][TPB];

    const int tid = threadIdx.x;
    const int b   = blockIdx.y;
    const int n   = blockIdx.x * TPB + tid;
    const int Mc  = (M < MAXM) ? M : MAXM;

    // ---- CDNA5 async DMA: stage this batch's gt boxes straight into LDS ----
    // GVS mode: saddr = uniform 64-bit base (SGPR pair), vaddr = per-lane byte
    // offset, vdst = LDS byte address (low 32 bits of flat LDS pointer).
    {
        const float* gbase = gt + (size_t)b * (size_t)M * 4;
        if (tid < Mc) {   // lanes 0..63 -> two fully-active wave32s
            unsigned lds_off = (unsigned)(size_t)&s_gt[tid];
            unsigned goff    = (unsigned)(tid * 16);
            asm volatile("global_load_async_to_lds_b128 %0, %1, %2"
                         :: "v"(lds_off), "v"(goff), "s"(gbase)
                         : "memory");
        }
        asm volatile("s_wait_asynccnt 0" ::: "memory");
        __syncthreads();
    }

    float focal_c = 0.0f, valid_c = 0.0f, sl1_c = 0.0f, pos_c = 0.0f;

    if (n < N) {
        // anchor cxcywh -> corners
        const float4 a = *(const float4*)(anchors + (size_t)n * 4);
        const float ax1 = a.x - a.z * 0.5f, ay1 = a.y - a.w * 0.5f;
        const float ax2 = a.x + a.z * 0.5f, ay2 = a.y + a.w * 0.5f;
        const float area_a = (ax2 - ax1) * (ay2 - ay1);

        // 64-way IoU max/argmax; carry matched box in registers (fuses the
        // take_along_axis gather). Strict '>' keeps first-occurrence argmax.
        float  best = -1.0f;
        float4 mbox = s_gt[0];
        #pragma unroll 8
        for (int m = 0; m < Mc; ++m) {
            const float4 g = s_gt[m];                      // LDS broadcast read
            const float lx = fmaxf(ax1, g.x), ly = fmaxf(ay1, g.y);
            const float rx = fminf(ax2, g.z), ry = fminf(ay2, g.w);
            const float w  = fmaxf(rx - lx, 0.0f);
            const float h  = fmaxf(ry - ly, 0.0f);
            const float inter  = w * h;
            const float area_g = (g.z - g.x) * (g.w - g.y);
            const float iou    = inter / (area_a + area_g - inter + EPS_);
            if (iou > best) { best = iou; mbox = g; }
        }

        const bool is_pos   = (best >= POS_T);
        const bool is_valid = is_pos || (best < NEG_T);

        // focal classification loss (valid anchors only)
        if (is_valid) {
            const float lg  = cls[(size_t)b * N + n];
            const float t   = is_pos ? 1.0f : 0.0f;
            const float ce  = fmaxf(lg, 0.0f) - lg * t + log1pf(expf(-fabsf(lg)));
            const float p   = 1.0f / (1.0f + expf(-lg));
            const float p_t = p * t + (1.0f - p) * (1.0f - t);
            const float a_t = ALPHA_ * t + (1.0f - ALPHA_) * (1.0f - t);
            const float om  = 1.0f - p_t;
            focal_c = a_t * om * om * ce;   // gamma = 2
            valid_c = 1.0f;
        }

        // smooth-L1 box loss (positive anchors only)
        if (is_pos) {
            pos_c = 1.0f;
            const float gcx = (mbox.x + mbox.z) * 0.5f;
            const float gcy = (mbox.y + mbox.w) * 0.5f;
            const float gw  =  mbox.z - mbox.x;
            const float gh  =  mbox.w - mbox.y;
            const float dx  = (gcx - a.x) / a.z;
            const float dy  = (gcy - a.y) / a.w;
            const float dw  = logf(gw / a.z + EPS_);
            const float dh  = logf(gh / a.w + EPS_);
            const float4 br = *(const float4*)(breg + ((size_t)b * N + n) * 4);
            const float d0 = fabsf(br.x - dx), d1 = fabsf(br.y - dy);
            const float d2 = fabsf(br.z - dw), d3 = fabsf(br.w - dh);
            sl1_c = (d0 < BETA_ ? 0.5f * d0 * d0 / BETA_ : d0 - 0.5f * BETA_)
                  + (d1 < BETA_ ? 0.5f * d1 * d1 / BETA_ : d1 - 0.5f * BETA_)
                  + (d2 < BETA_ ? 0.5f * d2 * d2 / BETA_ : d2 - 0.5f * BETA_)
                  + (d3 < BETA_ ? 0.5f * d3 * d3 / BETA_ : d3 - 0.5f * BETA_);
        }
    }

    // deterministic block tree reduction
    red[0][tid] = focal_c; red[1][tid] = valid_c;
    red[2][tid] = sl1_c;   red[3][tid] = pos_c;
    __syncthreads();
    for (int s = TPB / 2; s > 0; s >>= 1) {
        if (tid < s) {
            red[0][tid] += red[0][tid + s];
            red[1][tid] += red[1][tid + s];
            red[2][tid] += red[2][tid + s];
            red[3][tid] += red[3][tid + s];
        }
        __syncthreads();
    }
    if (tid == 0) {
        const size_t slot = (size_t)(blockIdx.y * gridDim.x + blockIdx.x) * 4;
        partials[slot + 0] = red[0][0];
        partials[slot + 1] = red[1][0];
        partials[slot + 2] = red[2][0];
        partials[slot + 3] = red[3][0];
    }
}

// Kernel 2: deterministic cross-block reduction + final loss scalars.
__global__ __launch_bounds__(256) void det_loss_finalize(
    const float* __restrict__ partials, int nblocks, float* __restrict__ out)
{
    __shared__ float red[4][256];
    const int tid = threadIdx.x;
    float a0 = 0.f, a1 = 0.f, a2 = 0.f, a3 = 0.f;
    for (int i = tid; i < nblocks; i += 256) {   // fixed per-thread order
        a0 += partials[(size_t)i * 4 + 0];
        a1 += partials[(size_t)i * 4 + 1];
        a2 += partials[(size_t)i * 4 + 2];
        a3 += partials[(size_t)i * 4 + 3];
    }
    red[0][tid] = a0; red[1][tid] = a1; red[2][tid] = a2; red[3][tid] = a3;
    __syncthreads();
    for (int s = 128; s > 0; s >>= 1) {
        if (tid < s) {
            red[0][tid] += red[0][tid + s];
            red[1][tid] += red[1][tid + s];
            red[2][tid] += red[2][tid + s];
            red[3][tid] += red[3][tid + s];
        }
        __syncthreads();
    }
    if (tid == 0) {
        const float fs = red[0][0], vs = red[1][0];
        const float ss = red[2][0], ps = red[3][0];
        const float cls_loss = fs / fmaxf(vs, 1.0f);
        const float npos     = ps * 4.0f;
        const float box_loss = (npos > 0.0f) ? (ss / fmaxf(npos, 1.0f)) : 0.0f;
        out[0] = cls_loss;
        out[1] = box_loss;
        out[2] = cls_loss + box_loss;   // BOX_W = 1
    }
}

extern "C" void kernel_launch(void* const* d_in, const int* in_sizes, int n_in,
                              void* d_out, int out_size, void* d_ws, size_t ws_size,
                              hipStream_t stream) {
    const float* cls  = (const float*)d_in[0];   // (B,N,1)
    const float* breg = (const float*)d_in[1];   // (B,N,4)
    const float* gt   = (const float*)d_in[2];   // (B,M,4)
    const float* anch = (const float*)d_in[3];   // (N,4)

    const int N = in_sizes[3] / 4;               // anchors: N*4
    const int B = in_sizes[0] / N;               // classification: B*N*1
    const int M = in_sizes[2] / (B * 4);         // gt: B*M*4

    const int gx = (N + TPB - 1) / TPB;
    dim3 grid(gx, B);
    float* partials = (float*)d_ws;              // gx*B*4 floats (16 KB)

    det_loss_main<<<grid, TPB, 0, stream>>>(cls, breg, gt, anch, partials, N, M);
    det_loss_finalize<<<1, 256, 0, stream>>>(partials, gx * B, (float*)d_out);
}